// EdgeCrossingLoss_57956288692604
// MI455X (gfx1250) — compile-verified
//
#include <hip/hip_runtime.h>
#include <hip/hip_bf16.h>

typedef __attribute__((ext_vector_type(2))) float v2f;
typedef __attribute__((ext_vector_type(8))) float v8f;

#define NF    16384
#define NV    8192
#define K_NB  19
#define TILE  16
#define NTILES (NF / TILE)
#define WAVES_PER_BLOCK 4
#define NTHREADS (WAVES_PER_BLOCK * 32)

// ---------------------------------------------------------------------------
// Kernel 1: centroids -> avec = (x,y,z,1)            [query / B side]
//                        bvec = (-2x,-2y,-2z,|c|^2)  [swept / A side]
// ---------------------------------------------------------------------------
__global__ void centroid_kernel(const float* __restrict__ vertices,
                                const int*   __restrict__ faces,
                                float4* __restrict__ avec,
                                float4* __restrict__ bvec) {
    int i = blockIdx.x * blockDim.x + threadIdx.x;
    if (i >= NF) return;
    int v0 = faces[3 * i + 0];
    int v1 = faces[3 * i + 1];
    int v2 = faces[3 * i + 2];
    const float inv3 = 1.0f / 3.0f;
    float cx = (vertices[3 * v0 + 0] + vertices[3 * v1 + 0] + vertices[3 * v2 + 0]) * inv3;
    float cy = (vertices[3 * v0 + 1] + vertices[3 * v1 + 1] + vertices[3 * v2 + 1]) * inv3;
    float cz = (vertices[3 * v0 + 2] + vertices[3 * v1 + 2] + vertices[3 * v2 + 2]) * inv3;
    float sq = cx * cx + cy * cy + cz * cz;
    avec[i] = make_float4(cx, cy, cz, 1.0f);
    bvec[i] = make_float4(-2.0f * cx, -2.0f * cy, -2.0f * cz, sq);
}

// ---------------------------------------------------------------------------
// LDS-resident top-19 insert.  One pass finds max (to replace) AND second-max
// so the new threshold is curmax' = max(second_max, c).  Called only on the
// rare accept path; the hot loop never touches the lists.
// ---------------------------------------------------------------------------
__device__ __forceinline__ void lds_insert(float c, int j,
                                           float* __restrict__ sb,
                                           int*   __restrict__ si,
                                           float& curmax) {
    float v[K_NB];
#pragma unroll
    for (int k = 0; k < K_NB; ++k) v[k] = sb[k];

    float m1 = v[0];                 // max
    float m2 = -__builtin_inff();    // second max
    int   p1 = 0;                    // argmax
#pragma unroll
    for (int k = 1; k < K_NB; ++k) {
        const bool g = (v[k] > m1);
        m2 = g ? m1 : fmaxf(m2, v[k]);
        m1 = g ? v[k] : m1;
        p1 = g ? k : p1;
    }
    sb[p1] = c;
    si[p1] = j;
    curmax = fmaxf(m2, c);
}

// ---------------------------------------------------------------------------
// Kernel 2: fused WMMA distance tiles + LDS-resident top-19 selection.
// Wave owns a fixed 16-column tile (queries) and sweeps all row tiles.
// D[m][n] = bvec_m . avec_n = -2 c_m.c_n + |c_m|^2 ; lane adds its |c_n|^2.
// WMMA D layout puts column N = lane&15, rows r + 8*(lane>>4) in each lane's
// registers -> candidates are lane-local.  Top-19 lists live in LDS so the
// hot loop carries almost no register state (kills phi-copy mov storms).
// ---------------------------------------------------------------------------
__global__ __launch_bounds__(NTHREADS)
void knn_kernel(const float4* __restrict__ avec,
                const float4* __restrict__ bvec,
                int* __restrict__ nb) {
    // [thread][k]; 19-dword lane stride is odd -> bank-conflict-free.
    __shared__ float sbest[NTHREADS][K_NB];
    __shared__ int   sidx [NTHREADS][K_NB];

    const int tid  = threadIdx.x;
    const int lane = tid & 31;               // wave32
    const int half = lane >> 4;              // row group (0..7 vs 8..15)
    const int l16  = lane & 15;

    // Wave-uniform query-tile index in an SGPR -> scalar diagonal branch.
    const int colTile = __builtin_amdgcn_readfirstlane(
        blockIdx.x * WAVES_PER_BLOCK + (tid >> 5));
    const int colbase = colTile * TILE;
    const int n = colbase + l16;             // query face owned by this lane pair

    float* const sb = sbest[tid];
    int*   const si = sidx[tid];
#pragma unroll
    for (int k = 0; k < K_NB; ++k) { sb[k] = __builtin_inff(); si[k] = 0; }

    // Fixed B fragment: column n = (x,y,z,1); lane half selects K {0,1}/{2,3}
    const float4 qa = avec[n];
    v2f bfrag;
    bfrag.x = half ? qa.z : qa.x;
    bfrag.y = half ? qa.w : qa.y;
    const float sqn = bvec[n].w;             // |c_n|^2

    float curmax = __builtin_inff();

    // Pipeline prologue: tile 0 in registers.
    float4 bm = bvec[l16];

    for (int mt = 0; mt < NTILES; ++mt) {
        const int mbase = mt * TILE;

        // Preload next tile (latency hidden behind WMMA + selection below).
        const int nidx = (mt + 1 < NTILES) ? (mbase + TILE + l16) : l16;
        const float4 bnext = bvec[nidx];
        __builtin_prefetch(&bvec[((mt + 2 < NTILES) ? (mbase + 2 * TILE) : 0) + l16],
                           0, 0);            // global_prefetch_b8, 2 tiles ahead

        // Swept A fragment: row m = mbase + l16 of (-2x,-2y,-2z,|c|^2)
        v2f afrag;
        afrag.x = half ? bm.z : bm.x;
        afrag.y = half ? bm.w : bm.y;

        v8f d = {};
        // 8 args: (neg_a, A, neg_b, B, c_mod, C, reuse_a, reuse_b)
        d = __builtin_amdgcn_wmma_f32_16x16x4_f32(false, afrag, false, bfrag,
                                                  (short)0, d, false, false);

        if (mt != colTile) {
            // Common case (scalar branch): no diagonal in this tile.
            // Tile-level gate: only descend if the best of 8 candidates wins.
            float tmin = d[0];
#pragma unroll
            for (int r = 1; r < 8; ++r) tmin = fminf(tmin, d[r]);
            if (tmin + sqn < curmax) {
#pragma unroll
                for (int r = 0; r < 8; ++r) {
                    const float c = d[r] + sqn;      // full d2(j, n)
                    if (c < curmax) {
                        lds_insert(c, mbase + r + 8 * half, sb, si, curmax);
                    }
                }
            }
        } else {
            // Diagonal tile (runs once per wave): exclude j == n.
#pragma unroll
            for (int r = 0; r < 8; ++r) {
                const int j = mbase + r + 8 * half;
                const float c = d[r] + sqn;
                if (j != n && c < curmax) {
                    lds_insert(c, j, sb, si, curmax);
                }
            }
        }

        bm = bnext;
    }

    // Merge the two half-lists for each query column (tid and tid^16, same
    // wave -> LDS ops are in program order; no barrier needed).  Snapshot the
    // partner list into registers BEFORE inserting, since both lanes mutate
    // their own lists during the merge.
    const int ptid = tid ^ 16;
    float pc[K_NB];
    int   pj[K_NB];
#pragma unroll
    for (int k = 0; k < K_NB; ++k) {
        pc[k] = sbest[ptid][k];
        pj[k] = sidx[ptid][k];
    }
#pragma unroll
    for (int k = 0; k < K_NB; ++k) {
        if (pc[k] < curmax) {
            lds_insert(pc[k], pj[k], sb, si, curmax);
        }
    }

    if (lane < 16) {
#pragma unroll
        for (int k = 0; k < K_NB; ++k) nb[n * K_NB + k] = si[k];
    }
}

// ---------------------------------------------------------------------------
// Kernel 3: edge-crossing tests, mirroring the reference math exactly
// (including exact float equality semantics), then wave32 reduce + atomicAdd.
// ---------------------------------------------------------------------------
__device__ __forceinline__ float3 f3sub(float3 a, float3 b) {
    return make_float3(a.x - b.x, a.y - b.y, a.z - b.z);
}
__device__ __forceinline__ float3 f3cross(float3 a, float3 b) {
    return make_float3(a.y * b.z - a.z * b.y,
                       a.z * b.x - a.x * b.z,
                       a.x * b.y - a.y * b.x);
}
__device__ __forceinline__ float f3dot(float3 a, float3 b) {
    return a.x * b.x + a.y * b.y + a.z * b.z;
}
__device__ __forceinline__ bool is_between(float3 p, float3 e0, float3 e1) {
    return (fabsf(p.x - e0.x) + fabsf(p.x - e1.x) == fabsf(e1.x - e0.x)) &&
           (fabsf(p.y - e0.y) + fabsf(p.y - e1.y) == fabsf(e1.y - e0.y)) &&
           (fabsf(p.z - e0.z) + fabsf(p.z - e1.z) == fabsf(e1.z - e0.z));
}

__global__ void crossing_kernel(const float* __restrict__ vertices,
                                const int*   __restrict__ faces,
                                const float* __restrict__ face_probs,
                                const int*   __restrict__ nb,
                                float* __restrict__ out) {
    const int i = blockIdx.x * blockDim.x + threadIdx.x;
    float part = 0.0f;

    if (i < NF) {
        float3 t[3];
#pragma unroll
        for (int a = 0; a < 3; ++a) {
            const int v = faces[3 * i + a];
            t[a] = make_float3(vertices[3 * v + 0], vertices[3 * v + 1], vertices[3 * v + 2]);
        }

        float cnt = 0.0f;
        for (int q = 0; q < K_NB; ++q) {
            const int j = nb[i * K_NB + q];
            float3 u[3];
#pragma unroll
            for (int a = 0; a < 3; ++a) {
                const int v = faces[3 * j + a];
                u[a] = make_float3(vertices[3 * v + 0], vertices[3 * v + 1], vertices[3 * v + 2]);
            }
#pragma unroll
            for (int a = 0; a < 3; ++a) {
                const float3 A = t[a];
                const float3 B = t[(a + 1) % 3];
                const float3 AB = f3sub(B, A);
#pragma unroll
                for (int c = 0; c < 3; ++c) {
                    const float3 C = u[c];
                    const float3 D = u[(c + 1) % 3];
                    const float3 CD = f3sub(D, C);
                    const float3 AC = f3sub(C, A);

                    const bool coplanar_ok = (f3dot(f3cross(AB, AC), CD) != 0.0f);
                    const float denom = f3dot(f3cross(AB, CD), CD);
                    const bool denom_ok = (denom != 0.0f);
                    const float3 negAC = make_float3(-AC.x, -AC.y, -AC.z);
                    const float num = f3dot(f3cross(negAC, CD), CD);
                    const float tpar = num / (denom_ok ? denom : 1.0f);
                    const bool t_ok = (tpar >= 0.0f) && (tpar <= 1.0f);
                    const float3 P = make_float3(A.x + tpar * AB.x,
                                                 A.y + tpar * AB.y,
                                                 A.z + tpar * AB.z);
                    const bool crossing = coplanar_ok && denom_ok && t_ok &&
                                          is_between(P, C, D) && is_between(P, A, B);
                    cnt += crossing ? 1.0f : 0.0f;
                }
            }
        }
        part = face_probs[i] * cnt;
    }

    // wave32 reduction, then one atomic per wave
#pragma unroll
    for (int off = 16; off > 0; off >>= 1) {
        part += __shfl_down(part, off, 32);
    }
    if ((threadIdx.x & 31) == 0) {
        atomicAdd(out, part);
    }
}

// ---------------------------------------------------------------------------
extern "C" void kernel_launch(void* const* d_in, const int* in_sizes, int n_in,
                              void* d_out, int out_size, void* d_ws, size_t ws_size,
                              hipStream_t stream) {
    const float* vertices   = (const float*)d_in[0]; // (V,3) f32
    const int*   faces      = (const int*)d_in[1];   // (F,3) i32
    const float* face_probs = (const float*)d_in[2]; // (F,)  f32
    float* out = (float*)d_out;                      // scalar f32

    char* ws = (char*)d_ws;
    float4* avec = (float4*)(ws);                                   // NF * 16 B
    float4* bvec = (float4*)(ws + (size_t)NF * sizeof(float4));     // NF * 16 B
    int*    nb   = (int*)   (ws + (size_t)2 * NF * sizeof(float4)); // NF*19 ints

    hipMemsetAsync(d_out, 0, sizeof(float), stream);

    centroid_kernel<<<NF / 256, 256, 0, stream>>>(vertices, faces, avec, bvec);
    knn_kernel<<<NTILES / WAVES_PER_BLOCK, NTHREADS, 0, stream>>>(avec, bvec, nb);
    crossing_kernel<<<NF / 256, 256, 0, stream>>>(vertices, faces, face_probs, nb, out);
}